// HybridEncoder_14903536517631
// MI455X (gfx1250) — compile-verified
//
#include <hip/hip_runtime.h>
#include <math.h>

#define N_NODES 50000
#define N_EDGES 800000
#define NNZ_H   800000
#define M_HE    10000
#define F_IN    128
#define F_HID   64
#define F_OUT   64

typedef float v2f __attribute__((ext_vector_type(2)));
typedef float v8f __attribute__((ext_vector_type(8)));

// ---------------------------------------------------------------------------
// C[N x 64] = A[N x K] * B[K x 64], all row-major fp32. N multiple of 16,
// K in {64,128}. One wave per 16x16 C tile via V_WMMA_F32_16X16X4_F32.
// Block = 128 threads = 4 waves -> 4 column tiles of one 16-row stripe.
// B is staged in LDS (K*64*4 <= 32KB).
// ---------------------------------------------------------------------------
__global__ __launch_bounds__(128)
void gemm_wmma_f32(const float* __restrict__ A, const float* __restrict__ B,
                   float* __restrict__ C, int K) {
  __shared__ float Bs[F_IN * 64];
  const int tid = threadIdx.x;
  const int kn  = K * 64;
  for (int i = tid; i < kn; i += 128) Bs[i] = B[i];
  __syncthreads();

  const int lane    = tid & 31;
  const int waveCol = (tid >> 5) * 16;   // 0,16,32,48
  const int m       = lane & 15;         // row-in-tile for A, col for B/C
  const int hh      = lane >> 4;         // half-wave select
  const long row0   = (long)blockIdx.x * 16;

  const float* Arow = A  + (row0 + m) * K;
  const float* Bp   = Bs + waveCol + m;

  v8f acc = {};
  for (int k0 = 0; k0 < K; k0 += 4) {
    const int ka = k0 + (hh << 1);
    v2f a, b;
    // A 16x4 f32 layout: VGPR0 = K{0|2}, VGPR1 = K{1|3} per half-wave
    a.x = Arow[ka];
    a.y = Arow[ka + 1];
    // B 4x16 layout mirrored: row K striped across lanes within a VGPR-half
    b.x = Bp[(long)ka * 64];
    b.y = Bp[(long)(ka + 1) * 64];
    acc = __builtin_amdgcn_wmma_f32_16x16x4_f32(
        false, a, false, b, (short)0, acc, false, false);
  }
  // C/D 16x16 f32: VGPR r holds M = r + 8*half, N = lane&15
  float* Cp = C + (row0 + (hh << 3)) * 64 + waveCol + m;
#pragma unroll
  for (int r = 0; r < 8; ++r) Cp[(long)r * 64] = acc[r];
}

// ---------------------------------------------------------------------------
// Elementwise / scatter helpers
// ---------------------------------------------------------------------------
__global__ void k_zero(float* __restrict__ p, long n) {
  long i = (long)blockIdx.x * blockDim.x + threadIdx.x;
  if (i < n) p[i] = 0.0f;
}

__global__ void k_count(const int* __restrict__ idx, float* __restrict__ deg, int n) {
  int i = blockIdx.x * blockDim.x + threadIdx.x;
  if (i < n) atomicAdd(&deg[idx[i]], 1.0f);
}

// dis[i] = rsqrt(edge_deg[i] + 1)   (self-loop contributes +1; always > 0)
__global__ void k_gcn_dis(float* __restrict__ d, int n) {
  int i = blockIdx.x * blockDim.x + threadIdx.x;
  if (i < n) d[i] = rsqrtf(d[i] + 1.0f);
}

// in-place: v -> (v > 0) ? 1/v : 0
__global__ void k_inv(float* __restrict__ d, long n) {
  long i = (long)blockIdx.x * blockDim.x + threadIdx.x;
  if (i < n) { float v = d[i]; d[i] = (v > 0.0f) ? (1.0f / v) : 0.0f; }
}

// agg[i,:] = dis[i]^2 * xw[i,:]   (self-loop term, also initializes agg)
__global__ void k_gcn_self(const float* __restrict__ xw, const float* __restrict__ dis,
                           float* __restrict__ agg, int n /* N*16 */) {
  int i = blockIdx.x * blockDim.x + threadIdx.x;
  if (i < n) {
    int node = i >> 4, q = (i & 15) << 2;
    float w = dis[node]; w *= w;
    const float4 v = *(const float4*)(xw + (long)node * 64 + q);
    *(float4*)(agg + (long)node * 64 + q) =
        make_float4(w * v.x, w * v.y, w * v.z, w * v.w);
  }
}

// agg[col,:] += dis[row]*dis[col]*xw[row,:]
__global__ void k_gcn_edge(const int* __restrict__ row, const int* __restrict__ col,
                           const float* __restrict__ dis, const float* __restrict__ xw,
                           float* __restrict__ agg, int n /* E*16 */) {
  int i = blockIdx.x * blockDim.x + threadIdx.x;
  if (i < n) {
    int e = i >> 4, q = (i & 15) << 2;
    int r = row[e], c = col[e];
    float w = dis[r] * dis[c];
    const float4 v = *(const float4*)(xw + (long)r * 64 + q);
    float* o = agg + (long)c * 64 + q;
    atomicAdd(o + 0, w * v.x);
    atomicAdd(o + 1, w * v.y);
    atomicAdd(o + 2, w * v.z);
    atomicAdd(o + 3, w * v.w);
  }
}

// out = agg + b  (optional ELU)
__global__ void k_bias_act(const float* __restrict__ agg, const float* __restrict__ b,
                           float* __restrict__ out, long n, int do_elu) {
  long i = (long)blockIdx.x * blockDim.x + threadIdx.x;
  if (i < n) {
    float v = agg[i] + b[i & 63];
    if (do_elu) v = (v > 0.0f) ? v : expm1f(v);
    out[i] = v;
  }
}

// ef[he,:] += xw[node,:]
__global__ void k_hyp_fwd(const int* __restrict__ nodei, const int* __restrict__ hei,
                          const float* __restrict__ xw, float* __restrict__ ef,
                          int n /* NNZ*16 */) {
  int i = blockIdx.x * blockDim.x + threadIdx.x;
  if (i < n) {
    int e = i >> 4, q = (i & 15) << 2;
    int nd = nodei[e], h = hei[e];
    const float4 v = *(const float4*)(xw + (long)nd * 64 + q);
    float* o = ef + (long)h * 64 + q;
    atomicAdd(o + 0, v.x);
    atomicAdd(o + 1, v.y);
    atomicAdd(o + 2, v.z);
    atomicAdd(o + 3, v.w);
  }
}

// agg[node,:] += Binv[he] * ef[he,:]
__global__ void k_hyp_back(const int* __restrict__ nodei, const int* __restrict__ hei,
                           const float* __restrict__ Binv, const float* __restrict__ ef,
                           float* __restrict__ agg, int n /* NNZ*16 */) {
  int i = blockIdx.x * blockDim.x + threadIdx.x;
  if (i < n) {
    int e = i >> 4, q = (i & 15) << 2;
    int nd = nodei[e], h = hei[e];
    float w = Binv[h];
    const float4 v = *(const float4*)(ef + (long)h * 64 + q);
    float* o = agg + (long)nd * 64 + q;
    atomicAdd(o + 0, w * v.x);
    atomicAdd(o + 1, w * v.y);
    atomicAdd(o + 2, w * v.z);
    atomicAdd(o + 3, w * v.w);
  }
}

// out = Dinv[node]*agg + b  (optional ELU)
__global__ void k_hyp_final(const float* __restrict__ agg, const float* __restrict__ Dinv,
                            const float* __restrict__ b, float* __restrict__ out,
                            long n, int do_elu) {
  long i = (long)blockIdx.x * blockDim.x + threadIdx.x;
  if (i < n) {
    long node = i >> 6;
    float v = Dinv[node] * agg[i] + b[i & 63];
    if (do_elu) v = (v > 0.0f) ? v : expm1f(v);
    out[i] = v;
  }
}

// z = sigmoid(gate)*x_s + (1-sigmoid(gate))*x_d
__global__ void k_fuse(const float* __restrict__ xs, const float* __restrict__ xd,
                       const float* __restrict__ gate, float* __restrict__ z, long n) {
  long i = (long)blockIdx.x * blockDim.x + threadIdx.x;
  if (i < n) {
    float a = 1.0f / (1.0f + __expf(-gate[0]));
    z[i] = a * xs[i] + (1.0f - a) * xd[i];
  }
}

// ---------------------------------------------------------------------------
extern "C" void kernel_launch(void* const* d_in, const int* in_sizes, int n_in,
                              void* d_out, int out_size, void* d_ws, size_t ws_size,
                              hipStream_t stream) {
  (void)in_sizes; (void)n_in; (void)out_size; (void)ws_size;

  const float* x   = (const float*)d_in[0];
  const int*   ei  = (const int*)d_in[1];     // [2,E]: row = ei, col = ei+E
  const int*   hyi = (const int*)d_in[2];     // [2,NNZ]: node = hyi, he = hyi+NNZ
  const float* g1w = (const float*)d_in[3];
  const float* g1b = (const float*)d_in[4];
  const float* g2w = (const float*)d_in[5];
  const float* g2b = (const float*)d_in[6];
  const float* h1w = (const float*)d_in[7];
  const float* h1b = (const float*)d_in[8];
  const float* h2w = (const float*)d_in[9];
  const float* h2b = (const float*)d_in[10];
  const float* gate = (const float*)d_in[11];

  float* out_z  = (float*)d_out;
  float* out_xs = out_z  + (long)N_NODES * F_OUT;
  float* out_xd = out_xs + (long)N_NODES * F_OUT;

  float* ws   = (float*)d_ws;
  float* xw   = ws;                              // N*64
  float* agg  = xw   + (long)N_NODES * 64;       // N*64
  float* h    = agg  + (long)N_NODES * 64;       // N*64
  float* dis  = h    + (long)N_NODES * 64;       // N
  float* Dinv = dis  + N_NODES;                  // N
  float* Binv = Dinv + N_NODES;                  // M (contiguous after Dinv)
  float* ef   = Binv + M_HE;                     // M*64

  const int T = 256;
  auto cd = [](long a, long b) { return (int)((a + b - 1) / b); };
  const long NF = (long)N_NODES * 64;
  const int  N16 = N_NODES * 16, E16 = N_EDGES * 16, Z16 = NNZ_H * 16;

  // ---------- GCN branch ----------
  k_zero   <<<cd(N_NODES, T), T, 0, stream>>>(dis, N_NODES);
  k_count  <<<cd(N_EDGES, T), T, 0, stream>>>(ei + N_EDGES, dis, N_EDGES);
  k_gcn_dis<<<cd(N_NODES, T), T, 0, stream>>>(dis, N_NODES);

  gemm_wmma_f32<<<N_NODES / 16, 128, 0, stream>>>(x, g1w, xw, F_IN);
  k_gcn_self<<<cd(N16, T), T, 0, stream>>>(xw, dis, agg, N16);
  k_gcn_edge<<<cd(E16, T), T, 0, stream>>>(ei, ei + N_EDGES, dis, xw, agg, E16);
  k_bias_act<<<cd(NF, T), T, 0, stream>>>(agg, g1b, h, NF, 1);

  gemm_wmma_f32<<<N_NODES / 16, 128, 0, stream>>>(h, g2w, xw, F_HID);
  k_gcn_self<<<cd(N16, T), T, 0, stream>>>(xw, dis, agg, N16);
  k_gcn_edge<<<cd(E16, T), T, 0, stream>>>(ei, ei + N_EDGES, dis, xw, agg, E16);
  k_bias_act<<<cd(NF, T), T, 0, stream>>>(agg, g2b, out_xs, NF, 0);

  // ---------- Hypergraph branch ----------
  k_zero <<<cd(N_NODES + M_HE, T), T, 0, stream>>>(Dinv, N_NODES + M_HE);
  k_count<<<cd(NNZ_H, T), T, 0, stream>>>(hyi,         Dinv, NNZ_H);
  k_count<<<cd(NNZ_H, T), T, 0, stream>>>(hyi + NNZ_H, Binv, NNZ_H);
  k_inv  <<<cd(N_NODES + M_HE, T), T, 0, stream>>>(Dinv, N_NODES + M_HE);

  gemm_wmma_f32<<<N_NODES / 16, 128, 0, stream>>>(x, h1w, xw, F_IN);
  k_zero    <<<cd((long)M_HE * 64, T), T, 0, stream>>>(ef, (long)M_HE * 64);
  k_hyp_fwd <<<cd(Z16, T), T, 0, stream>>>(hyi, hyi + NNZ_H, xw, ef, Z16);
  k_zero    <<<cd(NF, T), T, 0, stream>>>(agg, NF);
  k_hyp_back<<<cd(Z16, T), T, 0, stream>>>(hyi, hyi + NNZ_H, Binv, ef, agg, Z16);
  k_hyp_final<<<cd(NF, T), T, 0, stream>>>(agg, Dinv, h1b, h, NF, 1);

  gemm_wmma_f32<<<N_NODES / 16, 128, 0, stream>>>(h, h2w, xw, F_HID);
  k_zero    <<<cd((long)M_HE * 64, T), T, 0, stream>>>(ef, (long)M_HE * 64);
  k_hyp_fwd <<<cd(Z16, T), T, 0, stream>>>(hyi, hyi + NNZ_H, xw, ef, Z16);
  k_zero    <<<cd(NF, T), T, 0, stream>>>(agg, NF);
  k_hyp_back<<<cd(Z16, T), T, 0, stream>>>(hyi, hyi + NNZ_H, Binv, ef, agg, Z16);
  k_hyp_final<<<cd(NF, T), T, 0, stream>>>(agg, Dinv, h2b, out_xd, NF, 0);

  // ---------- Fusion ----------
  k_fuse<<<cd(NF, T), T, 0, stream>>>(out_xs, out_xd, gate, out_z, NF);
}